// SegModel_14010183320176
// MI455X (gfx1250) — compile-verified
//
#include <hip/hip_runtime.h>

#define NPTS 8192
#define NB   2
#define KNN  20
#define NEG  0.2f
#define TILE 1024

typedef float v2f __attribute__((ext_vector_type(2)));
typedef float v8f __attribute__((ext_vector_type(8)));

// ---------------------------------------------------------------------------
// KNN: one thread per query point; point cloud streamed through LDS tiles.
// key = 2*dot(q,p) - |p|^2  (row-constant -|q|^2 dropped; ordering identical
// to reference pairwise). Strict '>' insertion preserves top_k tie-breaking.
// ---------------------------------------------------------------------------
__global__ void knn_kernel(const float* __restrict__ x, int* __restrict__ idxb) {
  __shared__ float sx[TILE], sy[TILE], sz[TILE], sw[TILE];
  const int b = blockIdx.y;
  const int n = blockIdx.x * blockDim.x + threadIdx.x;
  const float* xb = x + (size_t)b * 6 * NPTS;
  const float qx = xb[n];
  const float qy = xb[NPTS + n];
  const float qz = xb[2 * NPTS + n];
  float best[KNN];
  int   bidx[KNN];
#pragma unroll
  for (int t = 0; t < KNN; ++t) { best[t] = -3.402823466e38f; bidx[t] = 0; }
  for (int t0 = 0; t0 < NPTS; t0 += TILE) {
    __syncthreads();
    for (int i = threadIdx.x; i < TILE; i += blockDim.x) {
      const int j = t0 + i;
      const float px = xb[j], py = xb[NPTS + j], pz = xb[2 * NPTS + j];
      sx[i] = px; sy[i] = py; sz[i] = pz;
      sw[i] = px * px + py * py + pz * pz;
    }
    __syncthreads();
#pragma unroll 4
    for (int i = 0; i < TILE; ++i) {
      const float dot = qx * sx[i] + qy * sy[i] + qz * sz[i];
      const float key = 2.f * dot - sw[i];
      if (key > best[KNN - 1]) {
        best[KNN - 1] = key; bidx[KNN - 1] = t0 + i;
#pragma unroll
        for (int t = KNN - 1; t > 0; --t) {
          if (best[t] > best[t - 1]) {
            const float tv = best[t]; best[t] = best[t - 1]; best[t - 1] = tv;
            const int   ti = bidx[t]; bidx[t] = bidx[t - 1]; bidx[t - 1] = ti;
          }
        }
      }
    }
  }
  int* op = idxb + ((size_t)b * NPTS + n) * KNN;
#pragma unroll
  for (int t = 0; t < KNN; ++t) op[t] = bidx[t];
}

// ---------------------------------------------------------------------------
// Derived weights: W1s = W1 with xyz cols * 10, zero-padded K 6->8.
// Wd{2,3}[o,c] = W[o,64+c] - W[o,c]   (center-term coefficients).
// ---------------------------------------------------------------------------
__global__ void prep_weights(const float* __restrict__ W1, const float* __restrict__ W2,
                             const float* __restrict__ W3, float* __restrict__ w1s,
                             float* __restrict__ wd2, float* __restrict__ wd3) {
  for (int i = threadIdx.x; i < 64 * 8; i += blockDim.x) {
    const int o = i >> 3, c = i & 7;
    float v = 0.f;
    if (c < 6) { v = W1[o * 6 + c]; if (c < 3) v *= 10.f; }
    w1s[i] = v;
  }
  for (int i = threadIdx.x; i < 64 * 64; i += blockDim.x) {
    const int o = i >> 6, c = i & 63;
    wd2[i] = W2[o * 128 + 64 + c] - W2[o * 128 + c];
    wd3[i] = W3[o * 128 + 64 + c] - W3[o * 128 + c];
  }
}

// ---------------------------------------------------------------------------
// Zero-pad x (6 rows) into xp (8 rows) per batch so the layer-1 GEMM needs no
// K predication anywhere near the WMMA.
// ---------------------------------------------------------------------------
__global__ void pad_x_kernel(const float* __restrict__ x, float* __restrict__ xp) {
  const int b = blockIdx.y;
  const int i = blockIdx.x * blockDim.x + threadIdx.x;  // 0 .. 8*NPTS-1
  const int c = i / NPTS;
  xp[(size_t)b * 8 * NPTS + i] = (c < 6) ? x[(size_t)b * 6 * NPTS + i] : 0.f;
}

// ---------------------------------------------------------------------------
// fp32 WMMA GEMM: D[b][M x NPTS] = A[M x K] * B[b][K x NPTS] (+bias, leaky).
// Compile-time K / bias / activation -> branchless body and epilogue.
// One wave computes TWO 16x16 tiles stacked in M (rows m0..m0+31), sharing
// each B fragment across two independent WMMAs (back-to-back XDL issue).
// A layout: lane holds row M=lane&15; K pair {0,1} (lanes 0-15) / {2,3} (16-31).
// B mirrors with columns. C/D: VGPR r -> row r + 8*(lane>=16), col lane&15.
// ---------------------------------------------------------------------------
template <int KVAL, bool HASBIAS, bool LEAKY>
__global__ void gemm_wmma_kernel(const float* __restrict__ A, int lda,
                                 const float* __restrict__ Bm, long bstride,
                                 float* __restrict__ D, long dstride,
                                 const float* __restrict__ bias) {
  const int lane = threadIdx.x & 31;
  const int wv   = threadIdx.x >> 5;
  const int col  = (blockIdx.x * 4 + wv) * 16 + (lane & 15);
  const int m0   = blockIdx.y * 32;
  const int kh   = (lane >> 4) << 1;  // 0 or 2
  const float* Ap0 = A + (m0 + (lane & 15)) * lda + kh;
  const float* Ap1 = Ap0 + 16 * lda;
  const float* Bp  = Bm + (long)blockIdx.z * bstride + (long)kh * NPTS + col;
  float*       Dp  = D  + (long)blockIdx.z * dstride + col;
  v8f acc0 = {0.f, 0.f, 0.f, 0.f, 0.f, 0.f, 0.f, 0.f};
  v8f acc1 = {0.f, 0.f, 0.f, 0.f, 0.f, 0.f, 0.f, 0.f};
#pragma unroll
  for (int k0 = 0; k0 < KVAL; k0 += 4) {
    v2f a0, a1, bv;
    a0.x = Ap0[k0];
    a0.y = Ap0[k0 + 1];
    a1.x = Ap1[k0];
    a1.y = Ap1[k0 + 1];
    bv.x = Bp[(long)k0 * NPTS];
    bv.y = Bp[(long)(k0 + 1) * NPTS];
    acc0 = __builtin_amdgcn_wmma_f32_16x16x4_f32(false, a0, false, bv,
                                                 (short)0, acc0, false, false);
    acc1 = __builtin_amdgcn_wmma_f32_16x16x4_f32(false, a1, false, bv,
                                                 (short)0, acc1, false, false);
  }
  const int rbase = m0 + ((lane >> 4) << 3);
#pragma unroll
  for (int r = 0; r < 8; ++r) {
    const int row0 = rbase + r;
    const int row1 = row0 + 16;
    float v0 = acc0[r];
    float v1 = acc1[r];
    if (HASBIAS) { v0 += bias[row0]; v1 += bias[row1]; }
    if (LEAKY) {
      v0 = (v0 > 0.f) ? v0 : NEG * v0;
      v1 = (v1 > 0.f) ? v1 : NEG * v1;
    }
    Dp[(long)row0 * NPTS] = v0;
    Dp[(long)row1 * NPTS] = v1;
  }
}

// ---------------------------------------------------------------------------
// Layer-1 combine: x1[o,n] = leaky( max_k g[o,idx(n,k)] + b1[o]
//                                   - 10 * sum_{c<3} W1[o,c]*mean_n[c] )
// out points at xcat slice 0; batch stride 192*NPTS.
// ---------------------------------------------------------------------------
__global__ void combine1_kernel(const int* __restrict__ idxb, const float* __restrict__ x,
                                const float* __restrict__ g, const float* __restrict__ W1,
                                const float* __restrict__ b1, float* __restrict__ out) {
  const int b = blockIdx.y;
  const int n = blockIdx.x * blockDim.x + threadIdx.x;
  const int* ip = idxb + ((size_t)b * NPTS + n) * KNN;
  const float* xb = x + (size_t)b * 6 * NPTS;
  float mx = 0.f, my = 0.f, mz = 0.f;
#pragma unroll 1
  for (int k = 0; k < KNN; ++k) {
    const int j = ip[k];
    mx += xb[j]; my += xb[NPTS + j]; mz += xb[2 * NPTS + j];
  }
  const float s = 1.f / (float)KNN;
  mx *= s; my *= s; mz *= s;
  float macc[64];
#pragma unroll
  for (int o = 0; o < 64; ++o) macc[o] = -3.402823466e38f;
#pragma unroll 1
  for (int k = 0; k < KNN; ++k) {
    const float* yp = g + (size_t)b * 64 * NPTS + ip[k];
#pragma unroll
    for (int o = 0; o < 64; ++o) macc[o] = fmaxf(macc[o], yp[o * NPTS]);
  }
  float* op = out + (size_t)b * 192 * NPTS + n;
#pragma unroll
  for (int o = 0; o < 64; ++o) {
    const float base = b1[o] - 10.f * (W1[o * 6] * mx + W1[o * 6 + 1] * my + W1[o * 6 + 2] * mz);
    float v = macc[o] + base;
    v = (v > 0.f) ? v : NEG * v;
    op[o * NPTS] = v;
  }
}

// ---------------------------------------------------------------------------
// Layers 2/3 combine: out[o,n] = leaky( max_k Y[o,idx(n,k)] + Z[o,n] )
// Z already contains Wd*x + bias. out points at xcat slice; stride 192*NPTS.
// ---------------------------------------------------------------------------
__global__ void combine23_kernel(const int* __restrict__ idxb, const float* __restrict__ Y,
                                 const float* __restrict__ Z, float* __restrict__ out) {
  const int b = blockIdx.y;
  const int n = blockIdx.x * blockDim.x + threadIdx.x;
  const int* ip = idxb + ((size_t)b * NPTS + n) * KNN;
  float macc[64];
#pragma unroll
  for (int o = 0; o < 64; ++o) macc[o] = -3.402823466e38f;
#pragma unroll 1
  for (int k = 0; k < KNN; ++k) {
    const float* yp = Y + (size_t)b * 64 * NPTS + ip[k];
#pragma unroll
    for (int o = 0; o < 64; ++o) macc[o] = fmaxf(macc[o], yp[o * NPTS]);
  }
  const float* zp = Z + (size_t)b * 64 * NPTS + n;
  float* op = out + (size_t)b * 192 * NPTS + n;
#pragma unroll
  for (int o = 0; o < 64; ++o) {
    float v = macc[o] + zp[o * NPTS];
    v = (v > 0.f) ? v : NEG * v;
    op[o * NPTS] = v;
  }
}

extern "C" void kernel_launch(void* const* d_in, const int* in_sizes, int n_in,
                              void* d_out, int out_size, void* d_ws, size_t ws_size,
                              hipStream_t stream) {
  const float* x  = (const float*)d_in[0];
  const float* W1 = (const float*)d_in[1];
  const float* b1 = (const float*)d_in[2];
  const float* W2 = (const float*)d_in[3];
  const float* b2 = (const float*)d_in[4];
  const float* W3 = (const float*)d_in[5];
  const float* b3 = (const float*)d_in[6];
  const float* W4 = (const float*)d_in[7];
  const float* b4 = (const float*)d_in[8];
  float* out = (float*)d_out;

  char* ws = (char*)d_ws;
  size_t off = 0;
  int*   idxb = (int*)(ws + off);   off += (size_t)NB * NPTS * KNN * sizeof(int);
  float* w1s  = (float*)(ws + off); off += 64 * 8 * sizeof(float);
  float* wd2  = (float*)(ws + off); off += 64 * 64 * sizeof(float);
  float* wd3  = (float*)(ws + off); off += 64 * 64 * sizeof(float);
  float* xpad = (float*)(ws + off); off += (size_t)NB * 8 * NPTS * sizeof(float);
  float* Ybuf = (float*)(ws + off); off += (size_t)NB * 64 * NPTS * sizeof(float);
  float* Zbuf = (float*)(ws + off); off += (size_t)NB * 64 * NPTS * sizeof(float);
  float* xcat = (float*)(ws + off); off += (size_t)NB * 192 * NPTS * sizeof(float);

  const long FS = (long)192 * NPTS;  // xcat batch stride
  const long YS = (long)64 * NPTS;   // Y/Z batch stride

  prep_weights<<<1, 256, 0, stream>>>(W1, W2, W3, w1s, wd2, wd3);
  pad_x_kernel<<<dim3(8 * NPTS / 256, NB), 256, 0, stream>>>(x, xpad);
  knn_kernel<<<dim3(NPTS / 256, NB), 256, 0, stream>>>(x, idxb);

  // layer 1: g = W1s(64x8) * xpad(8 x NPTS) -> gather-max combine -> xcat slice 0
  gemm_wmma_kernel<8, false, false><<<dim3(NPTS / 64, 2, NB), 128, 0, stream>>>(
      w1s, 8, xpad, (long)8 * NPTS, Ybuf, YS, nullptr);
  combine1_kernel<<<dim3(NPTS / 256, NB), 256, 0, stream>>>(idxb, x, Ybuf, W1, b1, xcat);

  // layer 2: Y = W2[:, :64]*x1 ; Z = Wd2*x1 + b2 ; combine -> xcat slice 1
  gemm_wmma_kernel<64, false, false><<<dim3(NPTS / 64, 2, NB), 128, 0, stream>>>(
      W2, 128, xcat, FS, Ybuf, YS, nullptr);
  gemm_wmma_kernel<64, true, false><<<dim3(NPTS / 64, 2, NB), 128, 0, stream>>>(
      wd2, 64, xcat, FS, Zbuf, YS, b2);
  combine23_kernel<<<dim3(NPTS / 256, NB), 256, 0, stream>>>(
      idxb, Ybuf, Zbuf, xcat + (size_t)64 * NPTS);

  // layer 3: same from slice 1 -> xcat slice 2
  gemm_wmma_kernel<64, false, false><<<dim3(NPTS / 64, 2, NB), 128, 0, stream>>>(
      W3, 128, xcat + (size_t)64 * NPTS, FS, Ybuf, YS, nullptr);
  gemm_wmma_kernel<64, true, false><<<dim3(NPTS / 64, 2, NB), 128, 0, stream>>>(
      wd3, 64, xcat + (size_t)64 * NPTS, FS, Zbuf, YS, b3);
  combine23_kernel<<<dim3(NPTS / 256, NB), 256, 0, stream>>>(
      idxb, Ybuf, Zbuf, xcat + (size_t)128 * NPTS);

  // final: out = leaky(W4(256x192) * xcat + b4)
  gemm_wmma_kernel<192, true, true><<<dim3(NPTS / 64, 8, NB), 128, 0, stream>>>(
      W4, 192, xcat, FS, out, (long)256 * NPTS, b4);
}